// CArap_42880953484258
// MI455X (gfx1250) — compile-verified
//
#include <hip/hip_runtime.h>
#include <stdint.h>

namespace {
constexpr int BQ     = 2;
constexpr int NVTX   = 100000;          // vertices per batch
constexpr int KN     = 16;              // max neighbors
constexpr int NVERT  = BQ * NVTX;       // 200000 total vertices
constexpr int TILE   = 256;             // threads per block == vertices per tile
constexpr int NTILES = (NVERT + TILE - 1) / TILE;   // 782
constexpr int GRID   = (NTILES + 1) / 2;            // 2 tiles/block -> prefetch overlaps
}

// ---------------- gfx1250 async-to-LDS primitives (ASYNCcnt path) ----------------
#if defined(__HIP_DEVICE_COMPILE__)
__device__ __forceinline__ void async_ld16(uint32_t lds, uint32_t goff, uint64_t base) {
  // GVS mode: mem = SGPR64 + VGPR32 offset ; VDST VGPR holds the LDS byte address
  asm volatile("global_load_async_to_lds_b128 %0, %1, %2"
               :: "v"(lds), "v"(goff), "s"(base) : "memory");
}
__device__ __forceinline__ void wait_async0() {
  asm volatile("s_wait_asynccnt 0" ::: "memory");
}
__device__ __forceinline__ void wait_ds0() {
  asm volatile("s_wait_dscnt 0" ::: "memory");
}
#else
__device__ __forceinline__ void async_ld16(uint32_t, uint32_t, uint64_t) {}
__device__ __forceinline__ void wait_async0() {}
__device__ __forceinline__ void wait_ds0() {}
#endif

// Stage one tile's neighbor indices (64B/thread) + weights (64B/thread) into LDS.
__device__ __forceinline__ void stage_tile(int t, int tid, uint32_t lds_idx, uint32_t lds_w,
                                           uint64_t g_nbl, uint64_t g_w) {
  const int v = t * TILE + tid;
  if (v < NVERT) {
    const uint32_t go = (uint32_t)v * 64u;     // row of 16 dwords
#pragma unroll
    for (int i = 0; i < 4; ++i) async_ld16(lds_idx + 16u * i, go + 16u * i, g_nbl);
#pragma unroll
    for (int i = 0; i < 4; ++i) async_ld16(lds_w + 16u * i, go + 16u * i, g_w);
  }
}

// One cyclic Jacobi rotation on symmetric A, accumulating eigenvectors into V.
__device__ __forceinline__ void jrot(float A[3][3], float V[3][3], int p, int q) {
  const float apq = A[p][q];
  if (apq * apq < 1e-30f) return;
  const float tau = (A[q][q] - A[p][p]) / (2.0f * apq);
  const float t   = copysignf(1.0f, tau) / (fabsf(tau) + sqrtf(1.0f + tau * tau));
  const float c   = 1.0f / sqrtf(1.0f + t * t);
  const float s   = t * c;
#pragma unroll
  for (int r = 0; r < 3; ++r) {            // A <- A*J
    const float arp = A[r][p], arq = A[r][q];
    A[r][p] = c * arp - s * arq;
    A[r][q] = s * arp + c * arq;
  }
#pragma unroll
  for (int r = 0; r < 3; ++r) {            // A <- J^T*A
    const float apr = A[p][r], aqr = A[q][r];
    A[p][r] = c * apr - s * aqr;
    A[q][r] = s * apr + c * aqr;
  }
#pragma unroll
  for (int r = 0; r < 3; ++r) {            // V <- V*J
    const float vrp = V[r][p], vrq = V[r][q];
    V[r][p] = c * vrp - s * vrq;
    V[r][q] = s * vrp + c * vrq;
  }
}

__global__ void __launch_bounds__(TILE)
carap_energy_kernel(const float* __restrict__ xyz1, const float* __restrict__ xyz2,
                    const int* __restrict__ nbl, const int* __restrict__ nnb,
                    const float* __restrict__ wmat, const float* __restrict__ alpha,
                    const float* __restrict__ area, const float* __restrict__ carap,
                    float* __restrict__ out)
{
  __shared__ int   s_idx[TILE * KN];   // 16 KB
  __shared__ float s_w[TILE * KN];     // 16 KB

  const int   tid = threadIdx.x;
  const float scl = alpha[0] * carap[0];

  // Wave-relative LDS byte offsets (low 32 bits of the flat shared address).
  const uint32_t lds_idx = (uint32_t)(uintptr_t)&s_idx[tid * KN];
  const uint32_t lds_w   = (uint32_t)(uintptr_t)&s_w[tid * KN];
  const uint64_t g_nbl   = (uint64_t)(uintptr_t)nbl;
  const uint64_t g_w     = (uint64_t)(uintptr_t)wmat;

  // Prologue: stage the first tile.
  stage_tile((int)blockIdx.x, tid, lds_idx, lds_w, g_nbl, g_w);

  for (int t = blockIdx.x; t < NTILES; t += gridDim.x) {
    const int v = t * TILE + tid;

    wait_async0();                     // tile t's idx/w landed in LDS

    // Copy my 16 indices + 16 weights into registers (each thread reads only
    // the slots its own async loads wrote -> no barrier needed).
    int   nb[KN];
    float wv[KN];
#pragma unroll
    for (int k = 0; k < KN; ++k) { nb[k] = s_idx[tid * KN + k]; wv[k] = s_w[tid * KN + k]; }
    wait_ds0();                        // LDS reads done -> safe to overwrite slots

    // Prefetch next tile into the same LDS slots, overlapped with all compute below.
    const int tn = t + (int)gridDim.x;
    if (tn < NTILES) stage_tile(tn, tid, lds_idx, lds_w, g_nbl, g_w);

    if (v < NVERT) {
      const int    b  = v / NVTX;
      const size_t vb = (size_t)v * 3;
      const float x1 = xyz1[vb], y1 = xyz1[vb + 1], z1 = xyz1[vb + 2];
      const float x2 = xyz2[vb], y2 = xyz2[vb + 1], z2 = xyz2[vb + 2];
      const int   nn = nnb[v];
      const float* bs1 = xyz1 + (size_t)b * (size_t)NVTX * 3;
      const float* bs2 = xyz2 + (size_t)b * (size_t)NVTX * 3;

      // ---- pass 1: S = sum_k w * e_rest (x) e_def  (3x3) ----
      float S[3][3] = {{0,0,0},{0,0,0},{0,0,0}};
#pragma unroll
      for (int k = 0; k < KN; ++k) {
        const float wk = (k < nn) ? wv[k] : 0.0f;
        const float* p1 = bs1 + (size_t)nb[k] * 3;
        const float* p2 = bs2 + (size_t)nb[k] * 3;
        const float ex = x1 - p1[0], ey = y1 - p1[1], ez = z1 - p1[2];
        const float fx = x2 - p2[0], fy = y2 - p2[1], fz = z2 - p2[2];
        const float a0 = wk * ex, a1 = wk * ey, a2 = wk * ez;
        S[0][0] += a0 * fx; S[0][1] += a0 * fy; S[0][2] += a0 * fz;
        S[1][0] += a1 * fx; S[1][1] += a1 * fy; S[1][2] += a1 * fz;
        S[2][0] += a2 * fx; S[2][1] += a2 * fy; S[2][2] += a2 * fz;
      }

      // ---- 3x3 SVD via Jacobi on S^T S ----
      float Am[3][3], Vm[3][3] = {{1,0,0},{0,1,0},{0,0,1}};
#pragma unroll
      for (int i = 0; i < 3; ++i)
#pragma unroll
        for (int j = 0; j < 3; ++j)
          Am[i][j] = S[0][i]*S[0][j] + S[1][i]*S[1][j] + S[2][i]*S[2][j];

#pragma unroll
      for (int sw = 0; sw < 4; ++sw) { jrot(Am, Vm, 0, 1); jrot(Am, Vm, 0, 2); jrot(Am, Vm, 1, 2); }

      // sort eigenpairs descending (flip applies to smallest singular value)
      float l0 = Am[0][0], l1 = Am[1][1], l2 = Am[2][2];
      if (l0 < l1) { float tt=l0; l0=l1; l1=tt;
#pragma unroll
        for (int r=0;r<3;++r){ float tv=Vm[r][0]; Vm[r][0]=Vm[r][1]; Vm[r][1]=tv; } }
      if (l0 < l2) { float tt=l0; l0=l2; l2=tt;
#pragma unroll
        for (int r=0;r<3;++r){ float tv=Vm[r][0]; Vm[r][0]=Vm[r][2]; Vm[r][2]=tv; } }
      if (l1 < l2) { float tt=l1; l1=l2; l2=tt;
#pragma unroll
        for (int r=0;r<3;++r){ float tv=Vm[r][1]; Vm[r][1]=Vm[r][2]; Vm[r][2]=tv; } }

      // U columns: u_m = normalize(S v_m)
      float U[3][3];
      float n2last = 0.0f;
#pragma unroll
      for (int m = 0; m < 3; ++m) {
        const float vx = Vm[0][m], vy = Vm[1][m], vz = Vm[2][m];
        const float ux = S[0][0]*vx + S[0][1]*vy + S[0][2]*vz;
        const float uy = S[1][0]*vx + S[1][1]*vy + S[1][2]*vz;
        const float uz = S[2][0]*vx + S[2][1]*vy + S[2][2]*vz;
        const float q2 = ux*ux + uy*uy + uz*uz;
        const float iv = 1.0f / sqrtf(q2 + 1e-30f);
        U[m][0] = ux*iv; U[m][1] = uy*iv; U[m][2] = uz*iv;
        if (m == 2) n2last = q2;
      }
      if (n2last < 1e-24f) {           // degenerate sigma_2: complete orthonormal frame
        U[2][0] = U[0][1]*U[1][2] - U[0][2]*U[1][1];
        U[2][1] = U[0][2]*U[1][0] - U[0][0]*U[1][2];
        U[2][2] = U[0][0]*U[1][1] - U[0][1]*U[1][0];
      }

      const float det =
            S[0][0]*(S[1][1]*S[2][2] - S[1][2]*S[2][1])
          - S[0][1]*(S[1][0]*S[2][2] - S[1][2]*S[2][0])
          + S[0][2]*(S[1][0]*S[2][1] - S[1][1]*S[2][0]);
      const float dsg = (det < 0.0f) ? -1.0f : 1.0f;

      // R = V diag(1,1,d) U^T
      float R[3][3];
#pragma unroll
      for (int i = 0; i < 3; ++i)
#pragma unroll
        for (int j = 0; j < 3; ++j)
          R[i][j] = Vm[i][0]*U[0][j] + Vm[i][1]*U[1][j] + dsg*Vm[i][2]*U[2][j];

      // ---- pass 2: per-axis weighted residual energy (re-gather; L0/L1 hits) ----
      float e0 = 0.0f, e1 = 0.0f, e2 = 0.0f;
#pragma unroll
      for (int k = 0; k < KN; ++k) {
        const float wk = (k < nn) ? wv[k] : 0.0f;
        const float* p1 = bs1 + (size_t)nb[k] * 3;
        const float* p2 = bs2 + (size_t)nb[k] * 3;
        const float ex = x1 - p1[0], ey = y1 - p1[1], ez = z1 - p1[2];
        const float fx = x2 - p2[0], fy = y2 - p2[1], fz = z2 - p2[2];
        const float gx = fx - (R[0][0]*ex + R[0][1]*ey + R[0][2]*ez);
        const float gy = fy - (R[1][0]*ex + R[1][1]*ey + R[1][2]*ez);
        const float gz = fz - (R[2][0]*ex + R[2][1]*ey + R[2][2]*ez);
        e0 += wk * gx * gx; e1 += wk * gy * gy; e2 += wk * gz * gz;
      }

      const float sc = scl * area[v];
      out[vb + 0] = sc * e0;
      out[vb + 1] = sc * e1;
      out[vb + 2] = sc * e2;
    }
  }
}

extern "C" void kernel_launch(void* const* d_in, const int* in_sizes, int n_in,
                              void* d_out, int out_size, void* d_ws, size_t ws_size,
                              hipStream_t stream) {
  (void)in_sizes; (void)n_in; (void)out_size; (void)d_ws; (void)ws_size;
  const float* xyz1  = (const float*)d_in[0];
  const float* xyz2  = (const float*)d_in[1];
  const int*   nbl   = (const int*)d_in[2];
  const int*   nnb   = (const int*)d_in[3];
  const float* wmat  = (const float*)d_in[4];
  const float* alpha = (const float*)d_in[5];
  const float* area  = (const float*)d_in[6];
  const float* carap = (const float*)d_in[7];
  float* out = (float*)d_out;

  hipLaunchKernelGGL(carap_energy_kernel, dim3(GRID), dim3(TILE), 0, stream,
                     xyz1, xyz2, nbl, nnb, wmat, alpha, area, carap, out);
}